// PFNLayer_Radar_logits_38646115730108
// MI455X (gfx1250) — compile-verified
//
#include <hip/hip_runtime.h>
#include <hip/hip_bf16.h>

typedef __attribute__((ext_vector_type(2))) float v2f;
typedef __attribute__((ext_vector_type(8))) float v8f;

// D = A(16x4 f32) * B(4x16 f32) + C(16x16 f32), wave32 WMMA
#define WMMA4(A, B, C) \
  __builtin_amdgcn_wmma_f32_16x16x4_f32(false, (A), false, (B), (short)0, (C), false, false)

// channel -> row-within-branch maps (compile-time folded; -1 = zero row)
__device__ constexpr int map_b0(int k) { return (k <= 2) ? k : ((k >= 5 && k <= 9) ? (k - 2) : -1); } // {0,1,2,5,6,7,8,9}
__device__ constexpr int map_b1(int k) { return (k == 3) ? 0 : ((k == 10) ? 1 : -1); }                 // {3,10}
__device__ constexpr int map_b2(int k) { return (k == 4) ? 0 : ((k == 11) ? 1 : -1); }                 // {4,11}
__device__ constexpr int map_b3(int k) { return (k >= 12) ? (k - 12) : -1; }                           // {12..15}

__device__ __forceinline__ float wval(const float* __restrict__ W, int r, int col) {
  return (r >= 0) ? W[r * 16 + col] : 0.0f;
}

// B fragment element for K-step j, slot s; BN scale folded into the weights
#define BV(W, MAP, j, s, INV) \
  ((hi ? wval((W), MAP(4*(j)+2+(s)), col) : wval((W), MAP(4*(j)+(s)), col)) * (INV))

// relu + running max, 1 v_max3_num_f32 per element (bias already in the accumulator)
#define REDUCE_BRANCH(C, MX)                                    \
  {                                                             \
    _Pragma("unroll") for (int r = 0; r < 8; ++r) {             \
      (MX) = fmaxf(fmaxf((C)[r], 0.0f), (MX));                  \
    }                                                           \
  }

__global__ __launch_bounds__(256) void pfn_radar_wmma_kernel(
    const float* __restrict__ in,      // (N, 32, 16) f32
    const float* __restrict__ W1,      // (8,16)
    const float* __restrict__ W2,      // (2,16)
    const float* __restrict__ W3,      // (2,16)
    const float* __restrict__ W4,      // (4,16)
    const float* __restrict__ gamma,   // (64,)
    const float* __restrict__ beta,    // (64,)
    const float* __restrict__ mean,    // (64,)
    const float* __restrict__ var,     // (64,)
    float* __restrict__ out,           // (N, 1, 64)
    int N)
{
  const int lane = threadIdx.x & 31;
  const int hi   = lane >> 4;   // high half of wave -> K offset +2, rows 0..15 repeated
  const int col  = lane & 15;   // output column / A row / memory row

  // ---- Eval-mode BN folded to per-channel scale+bias (channel = b*16 + col) ----
  const int ch0 = col, ch1 = 16 + col, ch2 = 32 + col, ch3 = 48 + col;
  const float inv0 = gamma[ch0] * rsqrtf(var[ch0] + 1e-3f);
  const float inv1 = gamma[ch1] * rsqrtf(var[ch1] + 1e-3f);
  const float inv2 = gamma[ch2] * rsqrtf(var[ch2] + 1e-3f);
  const float inv3 = gamma[ch3] * rsqrtf(var[ch3] + 1e-3f);
  const float bs0 = beta[ch0] - mean[ch0] * inv0;
  const float bs1 = beta[ch1] - mean[ch1] * inv1;
  const float bs2 = beta[ch2] - mean[ch2] * inv2;
  const float bs3 = beta[ch3] - mean[ch3] * inv3;

  // ---- Per-lane B fragments: zero-padded per-branch weights, BN scale pre-folded ----
  v2f B0_0 = { BV(W1, map_b0, 0, 0, inv0), BV(W1, map_b0, 0, 1, inv0) };
  v2f B0_1 = { BV(W1, map_b0, 1, 0, inv0), BV(W1, map_b0, 1, 1, inv0) };
  v2f B0_2 = { BV(W1, map_b0, 2, 0, inv0), BV(W1, map_b0, 2, 1, inv0) };
  v2f B1_0 = { BV(W2, map_b1, 0, 0, inv1), BV(W2, map_b1, 0, 1, inv1) };
  v2f B1_2 = { BV(W2, map_b1, 2, 0, inv1), BV(W2, map_b1, 2, 1, inv1) };
  v2f B2_1 = { BV(W3, map_b2, 1, 0, inv2), BV(W3, map_b2, 1, 1, inv2) };
  v2f B2_2 = { BV(W3, map_b2, 2, 0, inv2), BV(W3, map_b2, 2, 1, inv2) };
  v2f B3_3 = { BV(W4, map_b3, 3, 0, inv3), BV(W4, map_b3, 3, 1, inv3) };

  // ---- BN bias folded into the accumulator init (every C/D element of a lane is col-major bias) ----
  const v8f cini0 = { bs0, bs0, bs0, bs0, bs0, bs0, bs0, bs0 };
  const v8f cini1 = { bs1, bs1, bs1, bs1, bs1, bs1, bs1, bs1 };
  const v8f cini2 = { bs2, bs2, bs2, bs2, bs2, bs2, bs2, bs2 };
  const v8f cini3 = { bs3, bs3, bs3, bs3, bs3, bs3, bs3, bs3 };

  const int tid    = blockIdx.x * blockDim.x + threadIdx.x;
  const int wave   = tid >> 5;
  const int nwaves = (gridDim.x * blockDim.x) >> 5;

  for (int n = wave; n < N; n += nwaves) {       // wave-uniform -> EXEC all-1s at WMMAs
    const float* X = in + (size_t)n * 512;       // 32 points x 16 channels
    const float NEG = -3.402823466e38f;
    float mx0 = NEG, mx1 = NEG, mx2 = NEG, mx3 = NEG;

#pragma unroll
    for (int t = 0; t < 2; ++t) {                // points 0..15, then 16..31
      const float* Xt = X + t * 256 + col * 16 + 2 * hi;
      // A fragments for K-steps 0..3 (8B-aligned, streamed once -> non-temporal)
      v2f a0 = __builtin_nontemporal_load((const v2f*)(Xt + 0));
      v2f a1 = __builtin_nontemporal_load((const v2f*)(Xt + 4));
      v2f a2 = __builtin_nontemporal_load((const v2f*)(Xt + 8));
      v2f a3 = __builtin_nontemporal_load((const v2f*)(Xt + 12));

      v8f c0 = cini0, c1 = cini1, c2 = cini2, c3 = cini3;
      // structured-sparse K schedule: only steps with nonzero branch rows
      c0 = WMMA4(a0, B0_0, c0);
      c1 = WMMA4(a0, B1_0, c1);
      c0 = WMMA4(a1, B0_1, c0);
      c2 = WMMA4(a1, B2_1, c2);
      c0 = WMMA4(a2, B0_2, c0);
      c1 = WMMA4(a2, B1_2, c1);
      c2 = WMMA4(a2, B2_2, c2);
      c3 = WMMA4(a3, B3_3, c3);

      // ReLU + running max over this tile's 8 rows/lane (single max3 per element)
      REDUCE_BRANCH(c0, mx0);
      REDUCE_BRANCH(c1, mx1);
      REDUCE_BRANCH(c2, mx2);
      REDUCE_BRANCH(c3, mx3);
    }

    // combine the two lane-halves (rows r vs r+8) -> max over all 32 points
    mx0 = fmaxf(mx0, __shfl_xor(mx0, 16, 32));
    mx1 = fmaxf(mx1, __shfl_xor(mx1, 16, 32));
    mx2 = fmaxf(mx2, __shfl_xor(mx2, 16, 32));
    mx3 = fmaxf(mx3, __shfl_xor(mx3, 16, 32));

    if (lane < 16) {
      float* o = out + (size_t)n * 64 + col;
      __builtin_nontemporal_store(mx0, o);
      __builtin_nontemporal_store(mx1, o + 16);
      __builtin_nontemporal_store(mx2, o + 32);
      __builtin_nontemporal_store(mx3, o + 48);
    }
  }
}

extern "C" void kernel_launch(void* const* d_in, const int* in_sizes, int n_in,
                              void* d_out, int out_size, void* d_ws, size_t ws_size,
                              hipStream_t stream) {
  const float* in    = (const float*)d_in[0];
  const float* W1    = (const float*)d_in[1];
  const float* W2    = (const float*)d_in[2];
  const float* W3    = (const float*)d_in[3];
  const float* W4    = (const float*)d_in[4];
  const float* gamma = (const float*)d_in[5];
  const float* beta  = (const float*)d_in[6];
  const float* mean  = (const float*)d_in[7];
  const float* var   = (const float*)d_in[8];
  float* out = (float*)d_out;

  const int N = in_sizes[0] / (32 * 16);

  const int threads = 256;                 // 8 wave32 per block
  int blocks = 2048;                       // ~16k waves: saturates, ~9 pillars/wave
  int max_blocks = (N + 7) / 8;            // >=1 pillar per wave
  if (blocks > max_blocks) blocks = max_blocks;
  if (blocks < 1) blocks = 1;

  pfn_radar_wmma_kernel<<<blocks, threads, 0, stream>>>(
      in, W1, W2, W3, W4, gamma, beta, mean, var, out, N);
}